// SymbolicLayer_11622181503050
// MI455X (gfx1250) — compile-verified
//
#include <hip/hip_runtime.h>
#include <cmath>

// ---------------------------------------------------------------------------
// Problem constants (from reference): x[2048,256], layer_params[256,256,5]
// ---------------------------------------------------------------------------
#define B_ROWS 2048
#define I_DIM  256
#define O_DIM  256
#define K_FEAT (3 * I_DIM)   // blocked features: [x | x^2 | x^3] -> K = 768

typedef __attribute__((ext_vector_type(2))) float v2f;
typedef __attribute__((ext_vector_type(8))) float v8f;

// ---------------------------------------------------------------------------
// Shared per-(i,o) parameter decode (matches reference semantics exactly)
// ---------------------------------------------------------------------------
__device__ __forceinline__ void decode_params(const float* __restrict__ p5,
                                              float& a, float& b, float& c,
                                              float& d, int& idx) {
    a = p5[0]; b = p5[1]; c = p5[2]; d = p5[3];
    // jnp.round = round-half-to-even = rintf under default rounding mode
    int r = (int)rintf(p5[4]);
    idx = r < 0 ? 0 : (r > 5 ? 5 : r);
}

__device__ __forceinline__ float poly_const_term(float a, float b, float c, int idx) {
    // constant (x^0) coefficient of c*(a x + b)^n for idx in {3,4,5}
    if (idx == 3) return c * b;
    if (idx == 4) return c * b * b;
    if (idx == 5) return c * b * b * b;
    return 0.0f;
}

// ---------------------------------------------------------------------------
// Pass 1a: build packed transcendental table TP[i*O+o] = {a, b, c_t, code}
//          and BLOCKED polynomial coefficient matrix:
//          W[(p*I + i)*O + o] = coefficient of x_i^(p+1) for output o
// grid = I_DIM blocks, block = O_DIM threads
// ---------------------------------------------------------------------------
__global__ void symb_prep_kernel(const float* __restrict__ lp,
                                 float4* __restrict__ tp,
                                 float* __restrict__ W) {
    const int i = blockIdx.x;
    const int o = threadIdx.x;
    float a, b, c, d, w1 = 0.f, w2 = 0.f, w3 = 0.f;
    int idx;
    decode_params(lp + ((size_t)i * O_DIM + o) * 5, a, b, c, d, idx);

    float c_t;      // multiplier used by the transcendental (VALU) pass
    int   code;     // 0=sin 1=cos 2=exp 3=identity (poly entries: c_t=0)
    if (idx < 3) {
        c_t = c; code = idx;
    } else {
        c_t = 0.0f; code = 3;
        if (idx == 3) {           // c*(a x + b)         = (c a) x + c b
            w1 = c * a;
        } else if (idx == 4) {    // c*(a x + b)^2
            w1 = 2.0f * c * a * b;
            w2 = c * a * a;
        } else {                  // c*(a x + b)^3
            w1 = 3.0f * c * a * b * b;
            w2 = 3.0f * c * a * a * b;
            w3 = c * a * a * a;
        }
    }
    tp[(size_t)i * O_DIM + o] = make_float4(a, b, c_t, (float)code);
    // blocked layout: power p in outer dimension (uniform per K-stretch)
    W[((size_t)(0 * I_DIM + i)) * O_DIM + o] = w1;
    W[((size_t)(1 * I_DIM + i)) * O_DIM + o] = w2;
    W[((size_t)(2 * I_DIM + i)) * O_DIM + o] = w3;
}

// ---------------------------------------------------------------------------
// Pass 1b: K0[o] = sum_i ( d[i,o] + poly-constant-term(i,o) )
// grid = O_DIM blocks (one per o), block = I_DIM threads (one per i)
// ---------------------------------------------------------------------------
__global__ void symb_k0_kernel(const float* __restrict__ lp,
                               float* __restrict__ K0) {
    const int o = blockIdx.x;
    const int i = threadIdx.x;
    float a, b, c, d;
    int idx;
    decode_params(lp + ((size_t)i * O_DIM + o) * 5, a, b, c, d, idx);
    float v = d + poly_const_term(a, b, c, idx);

    __shared__ float red[I_DIM];
    red[i] = v;
    __syncthreads();
    #pragma unroll
    for (int s = I_DIM / 2; s > 0; s >>= 1) {
        if (i < s) red[i] += red[i + s];
        __syncthreads();
    }
    if (i == 0) K0[o] = red[0];
}

// ---------------------------------------------------------------------------
// Pass 2: transcendental reduction (VALU/trans pipe)
// out[b,o] = K0[o] + sum_i c_t * f_code(a*x[b,i] + b)
// grid = B_ROWS blocks (one batch row), block = O_DIM threads (one per o)
// x row staged in LDS -> broadcast reads, TP reads are coalesced 16B.
// ---------------------------------------------------------------------------
__global__ void symb_trans_kernel(const float* __restrict__ x,
                                  const float4* __restrict__ tp,
                                  const float* __restrict__ K0,
                                  float* __restrict__ out) {
    const int b = blockIdx.x;
    const int o = threadIdx.x;

    __shared__ float xs[I_DIM];
    xs[o] = x[(size_t)b * I_DIM + o];
    __syncthreads();

    float acc = 0.0f;
    #pragma unroll 4
    for (int i = 0; i < I_DIM; ++i) {
        float4 t = tp[(size_t)i * O_DIM + o];
        float  xv = xs[i];
        float  lt = fmaf(t.x, xv, t.y);
        int    code = (int)t.w;
        // cos(t) = sin(t + pi/2): one v_sin covers both sin and cos
        float arg = (code == 1) ? lt + 1.5707963267948966f : lt;
        float sc  = __sinf(arg);           // -> v_sin_f32
        float ex  = __expf(lt);            // -> v_exp_f32
        float v   = (code <= 1) ? sc : ((code == 2) ? ex : lt);
        acc = fmaf(t.z, v, acc);
    }
    out[(size_t)b * O_DIM + o] = acc + K0[o];
}

// ---------------------------------------------------------------------------
// Pass 3: polynomial-branch GEMM on the matrix pipe
// out[16x16 tile] += A(16x4 f32) x B(4x16 f32), K accumulated over 768
// Blocked K: for p in {0,1,2}: A[m][k] = x[m][k]^(p+1), B = W block p.
// 4 waves per block, one wave32 per output tile; no divergence -> EXEC
// stays all-ones across every v_wmma (ISA 7.12 requirement).
// Fragment layouts per CDNA5 ISA 7.12.2 (32-bit A 16x4, 32-bit C/D 16x16).
// ---------------------------------------------------------------------------
__global__ void symb_wmma_kernel(const float* __restrict__ x,
                                 const float* __restrict__ W,
                                 float* __restrict__ out) {
    const int wave = threadIdx.x >> 5;
    const int tile = blockIdx.x * 4 + wave;      // (B/16)*(O/16) tiles total
    const int tn   = tile % (O_DIM / 16);
    const int tm   = tile / (O_DIM / 16);
    const int lane = threadIdx.x & 31;           // wave32
    const int half = lane >> 4;                  // 0: lanes 0-15, 1: lanes 16-31
    const int l    = lane & 15;
    const int m0   = tm * 16;
    const int n0   = tn * 16;

    // C/D fragment: VGPR r holds row (m0 + half*8 + r), column (n0 + l)
    const int crow = m0 + half * 8;
    const int ccol = n0 + l;
    v8f c;
    #pragma unroll
    for (int r = 0; r < 8; ++r)
        c[r] = out[(size_t)(crow + r) * O_DIM + ccol];

    const float*  xrow = x + (size_t)(m0 + l) * I_DIM;
    const float2* xrow2 = (const float2*)xrow;   // kk is always even -> 8B aligned

    #pragma unroll
    for (int p = 0; p < 3; ++p) {                // uniform power selector
        const float* Wp = W + (size_t)p * I_DIM * O_DIM + (n0 + l);
        #pragma unroll 2
        for (int k = 0; k < I_DIM; k += 4) {
            const int kk = k + (half << 1);      // this lane's K pair: kk, kk+1
            // A fragment: one global_load_b64 per lane, power applied uniformly
            float2 xp = xrow2[kk >> 1];
            v2f a, bm;
            float f0 = xp.x, f1 = xp.y;
            if (p >= 1) { f0 *= xp.x; f1 *= xp.y; }     // x^2
            if (p == 2) { f0 *= xp.x; f1 *= xp.y; }     // x^3
            a[0] = f0;
            a[1] = f1;
            // B fragment: VGPR0 = W[p][kk][n0+l], VGPR1 = W[p][kk+1][n0+l]
            bm[0] = Wp[(size_t)kk * O_DIM];
            bm[1] = Wp[(size_t)(kk + 1) * O_DIM];
            // D = A x B + C (f32 exact, same precision class as reference)
            c = __builtin_amdgcn_wmma_f32_16x16x4_f32(
                    /*neg_a=*/false, a, /*neg_b=*/false, bm,
                    /*c_mod=*/(short)0, c, /*reuse_a=*/false, /*reuse_b=*/false);
        }
    }

    #pragma unroll
    for (int r = 0; r < 8; ++r)
        out[(size_t)(crow + r) * O_DIM + ccol] = c[r];
}

// ---------------------------------------------------------------------------
// Fallback: fully scalar path (used only if workspace is too small)
// ---------------------------------------------------------------------------
__global__ void symb_fallback_kernel(const float* __restrict__ x,
                                     const float* __restrict__ lp,
                                     float* __restrict__ out) {
    const int b = blockIdx.x;
    const int o = threadIdx.x;
    __shared__ float xs[I_DIM];
    xs[o] = x[(size_t)b * I_DIM + o];
    __syncthreads();

    float acc = 0.0f;
    for (int i = 0; i < I_DIM; ++i) {
        float a, bb, cc, d;
        int idx;
        decode_params(lp + ((size_t)i * O_DIM + o) * 5, a, bb, cc, d, idx);
        float lt = fmaf(a, xs[i], bb);
        float v;
        switch (idx) {
            case 0: v = __sinf(lt); break;
            case 1: v = __cosf(lt); break;
            case 2: v = __expf(lt); break;
            case 3: v = lt; break;
            case 4: v = lt * lt; break;
            default: v = lt * lt * lt; break;
        }
        acc = fmaf(cc, v, acc) + d;
    }
    out[(size_t)b * O_DIM + o] = acc;
}

// ---------------------------------------------------------------------------
// Launcher
// ---------------------------------------------------------------------------
extern "C" void kernel_launch(void* const* d_in, const int* in_sizes, int n_in,
                              void* d_out, int out_size, void* d_ws, size_t ws_size,
                              hipStream_t stream) {
    const float* x  = (const float*)d_in[0];   // [2048, 256]
    const float* lp = (const float*)d_in[1];   // [256, 256, 5]
    float* out = (float*)d_out;                // [2048, 256]

    const size_t tp_bytes = (size_t)I_DIM * O_DIM * sizeof(float4); // 1 MB
    const size_t w_bytes  = (size_t)K_FEAT * O_DIM * sizeof(float); // 768 KB
    const size_t k0_bytes = (size_t)O_DIM * sizeof(float);          // 1 KB

    if (ws_size < tp_bytes + w_bytes + k0_bytes) {
        symb_fallback_kernel<<<B_ROWS, O_DIM, 0, stream>>>(x, lp, out);
        return;
    }

    char* ws = (char*)d_ws;
    float4* tp = (float4*)ws;
    float*  W  = (float*)(ws + tp_bytes);
    float*  K0 = (float*)(ws + tp_bytes + w_bytes);

    // Pass 1: decode params once (reused by all 2048 batch rows)
    symb_prep_kernel<<<I_DIM, O_DIM, 0, stream>>>(lp, tp, W);
    symb_k0_kernel<<<O_DIM, I_DIM, 0, stream>>>(lp, K0);

    // Pass 2: transcendental branches on the VALU/trans pipe
    symb_trans_kernel<<<B_ROWS, O_DIM, 0, stream>>>(x, tp, K0, out);

    // Pass 3: polynomial branches as f32 WMMA GEMM: out += [x|x^2|x^3] * W
    const int tiles = (B_ROWS / 16) * (O_DIM / 16);   // 2048
    symb_wmma_kernel<<<tiles / 4, 128, 0, stream>>>(x, W, out);
}